// PLIFNodeHetero_11596411699418
// MI455X (gfx1250) — compile-verified
//
#include <hip/hip_runtime.h>
#include <cstdint>

// Problem constants (from reference): x is [B=4, N=1024, T=128, D=128] fp32.
#define T_LEN 128
#define D_LEN 128
#define WAVES_PER_BLOCK 8
#define DEPTH 8            // async pipeline depth (time steps in flight per wave)

typedef __attribute__((ext_vector_type(4))) float float4v;
typedef int v4i __attribute__((vector_size(4 * sizeof(int))));  // matches builtin pointee

#if defined(__has_builtin)
#if __has_builtin(__builtin_amdgcn_global_load_async_to_lds_b128) && \
    __has_builtin(__builtin_amdgcn_s_wait_asynccnt)
#define USE_ASYNC_LDS 1
#endif
#endif

#ifndef USE_ASYNC_LDS
#define USE_ASYNC_LDS 0
#endif

#if USE_ASYNC_LDS
__device__ __forceinline__ void async_cp16(const float* gsrc, float* ldst) {
  // 16B per lane, memory -> LDS, tracked by ASYNCcnt.
  __builtin_amdgcn_global_load_async_to_lds_b128(
      (__attribute__((address_space(1))) v4i*)gsrc,
      (__attribute__((address_space(3))) v4i*)ldst,
      /*offset=*/0, /*cpol=*/0);
}
#endif

__device__ __forceinline__ float sigmoidf_(float z) {
  return 1.0f / (1.0f + __expf(-z));
}

__global__ __launch_bounds__(256)
void plif_scan_kernel(const float* __restrict__ x,
                      const float* __restrict__ decay,
                      const float* __restrict__ v_th,
                      float* __restrict__ out,
                      int rows) {
#if USE_ASYNC_LDS
  __shared__ float tile[WAVES_PER_BLOCK][DEPTH][D_LEN];  // 32 KB
#endif
  const int lane = threadIdx.x & 31;
  const int wave = threadIdx.x >> 5;
  const int row  = blockIdx.x * WAVES_PER_BLOCK + wave;  // one wave per (b,n) row
  if (row >= rows) return;

  const int f = lane * 4;  // 4 consecutive features per lane

  // Per-feature parameters (uniform across the whole sequence).
  const float4v dec_raw = *(const float4v*)(decay + f);
  const float4v vth_raw = *(const float4v*)(v_th + f);
  float4v d4, th4;
#pragma unroll
  for (int i = 0; i < 4; ++i) {
    d4[i]  = sigmoidf_(dec_raw[i]);
    th4[i] = sigmoidf_(vth_raw[i]) + 0.1f;
  }

  const float* __restrict__ xrow = x   + (size_t)row * (T_LEN * D_LEN) + f;
  float*       __restrict__ orow = out + (size_t)row * (T_LEN * D_LEN) + f;

  float4v v = {0.0f, 0.0f, 0.0f, 0.0f};

#if USE_ASYNC_LDS
  // ---- prologue: fill the DEPTH-deep async pipeline ----
#pragma unroll
  for (int s = 0; s < DEPTH; ++s)
    async_cp16(xrow + s * D_LEN, &tile[wave][s][f]);

  // ---- steady state: t in [0, T-DEPTH) keeps DEPTH loads in flight ----
  for (int t = 0; t < T_LEN - DEPTH; ++t) {
    __builtin_amdgcn_s_wait_asynccnt(DEPTH - 1);  // oldest (stage t) landed
    const int stage = t & (DEPTH - 1);
    float4v xv = *(const float4v*)&tile[wave][stage][f];  // ds_load_b128
    // Close LDS RAW/WAR hazard: the refill below rewrites this stage.
    asm volatile("s_wait_dscnt 0x0" ::: "memory");
    async_cp16(xrow + (t + DEPTH) * D_LEN, &tile[wave][stage][f]);

    float4v s4;
#pragma unroll
    for (int i = 0; i < 4; ++i) {
      float vv = v[i] * d4[i] + xv[i];        // leaky integrate
      float sp = (vv > th4[i]) ? 1.0f : 0.0f; // fire
      v[i] = (sp != 0.0f) ? 0.0f : vv;        // hard reset
      s4[i] = sp;
    }
    *(float4v*)(orow + t * D_LEN) = s4;       // global_store_b128
  }

  // ---- epilogue: drain remaining DEPTH stages ----
  __builtin_amdgcn_s_wait_asynccnt(0);
#pragma unroll
  for (int k = 0; k < DEPTH; ++k) {
    const int t = T_LEN - DEPTH + k;
    float4v xv = *(const float4v*)&tile[wave][t & (DEPTH - 1)][f];
    float4v s4;
#pragma unroll
    for (int i = 0; i < 4; ++i) {
      float vv = v[i] * d4[i] + xv[i];
      float sp = (vv > th4[i]) ? 1.0f : 0.0f;
      v[i] = (sp != 0.0f) ? 0.0f : vv;
      s4[i] = sp;
    }
    *(float4v*)(orow + t * D_LEN) = s4;
  }
#else
  // Fallback: direct vectorized global loads + gfx1250 prefetch.
#pragma unroll 4
  for (int t = 0; t < T_LEN; ++t) {
    if (t + 16 < T_LEN)
      __builtin_prefetch(xrow + (t + 16) * D_LEN, 0, 0);  // global_prefetch_b8
    float4v xv = *(const float4v*)(xrow + t * D_LEN);
    float4v s4;
#pragma unroll
    for (int i = 0; i < 4; ++i) {
      float vv = v[i] * d4[i] + xv[i];
      float sp = (vv > th4[i]) ? 1.0f : 0.0f;
      v[i] = (sp != 0.0f) ? 0.0f : vv;
      s4[i] = sp;
    }
    *(float4v*)(orow + t * D_LEN) = s4;
  }
#endif
}

extern "C" void kernel_launch(void* const* d_in, const int* in_sizes, int n_in,
                              void* d_out, int out_size, void* d_ws, size_t ws_size,
                              hipStream_t stream) {
  (void)n_in; (void)out_size; (void)d_ws; (void)ws_size;
  const float* x     = (const float*)d_in[0];
  const float* decay = (const float*)d_in[1];
  const float* v_th  = (const float*)d_in[2];
  float* out = (float*)d_out;

  const int rows = in_sizes[0] / (T_LEN * D_LEN);  // B*N = 4096
  const int blocks = (rows + WAVES_PER_BLOCK - 1) / WAVES_PER_BLOCK;  // 512
  plif_scan_kernel<<<blocks, 256, 0, stream>>>(x, decay, v_th, out, rows);
}